// KGEM_69097433858560
// MI455X (gfx1250) — compile-verified
//
#include <hip/hip_runtime.h>
#include <hip/hip_bf16.h>
#include <cstdint>

// Problem constants (from reference)
#define N_S   64        // samples
#define N_E   10000     // entities
#define DD    256       // feature dim
#define KSEL  10        // top-k
#define NPAD  10048     // entities padded to multiple of 64
#define BN    64        // entity tile
#define BK    32        // k-chunk
#define NCHUNK (DD / BK)       // 8
#define LDT   36        // LDS row stride in floats (mult of 4 -> 16B aligned rows)
#define ASYNC_PER_CHUNK 4      // async b128 instructions issued per wave per chunk

// ---------------- CDNA5 async global->LDS (ASYNCcnt) helpers ----------------
typedef int vint4 __attribute__((vector_size(16)));
typedef __attribute__((address_space(1))) vint4* g_vint4_p;
typedef __attribute__((address_space(3))) vint4* l_vint4_p;

__device__ __forceinline__ void async_b128(void* lds, const void* g) {
  // Build addrspace pointers via integer casts: a generic LDS pointer's low 32
  // bits are the LDS byte offset (aperture encoding); global generic == as(1).
  unsigned long long gaddr = (unsigned long long)g;
  unsigned long long loff  = (unsigned)(unsigned long long)lds;
#if __has_builtin(__builtin_amdgcn_global_load_async_to_lds_b128)
  __builtin_amdgcn_global_load_async_to_lds_b128(
      (g_vint4_p)gaddr, (l_vint4_p)loff, 0, 0);
#else
  asm volatile("global_load_async_to_lds_b128 %0, %1, off"
               :: "v"((unsigned)loff), "v"(gaddr)
               : "memory");
#endif
}

#if __has_builtin(__builtin_amdgcn_s_wait_asynccnt)
#define ASYNC_WAIT(n) __builtin_amdgcn_s_wait_asynccnt(n)
#else
#define ASYNC_WAIT(n) asm volatile("s_wait_asynccnt %0" :: "i"(n) : "memory")
#endif

// ------- Kernel A (first in file so its asm shows in the disasm snippet):
//         min-plus semiring GEMM with async-LDS double buffering ----
__global__ __launch_bounds__(256) void jaccard_minplus(
    const float* __restrict__ Q, const float* __restrict__ E,
    const float* __restrict__ qsum, const float* __restrict__ esum,
    float* __restrict__ jacc) {
  __shared__ float As[2][64 * LDT];   // queries tile  [m][k]
  __shared__ float Bs[2][64 * LDT];   // entities tile [n][k]
  const int tid = threadIdx.x;
  const int gn0 = blockIdx.x * BN;
  const int m0  = (tid >> 4) << 2;    // 4 query rows per thread
  const int n0  = (tid & 15) << 2;    // 4 entity cols per thread

  float acc[4][4];
#pragma unroll
  for (int i = 0; i < 4; ++i)
#pragma unroll
    for (int j = 0; j < 4; ++j) acc[i][j] = 0.f;

  auto issue = [&](int b, int kt) {
#pragma unroll
    for (int p = 0; p < 2; ++p) {
      int v   = tid + 256 * p;   // vec4 id 0..511
      int row = v >> 3;          // 0..63
      int c4  = v & 7;           // 0..7 vec4 within 32-float chunk
      int col = kt * BK + (c4 << 2);
      async_b128(&As[b][row * LDT + (c4 << 2)], Q + (size_t)row * DD + col);
      int er = gn0 + row; er = er < N_E ? er : (N_E - 1);  // clamp tail (values unused)
      async_b128(&Bs[b][row * LDT + (c4 << 2)], E + (size_t)er * DD + col);
    }
  };

  issue(0, 0);
  int buf = 0;
  for (int kt = 0; kt < NCHUNK; ++kt) {
    if (kt + 1 < NCHUNK) { issue(buf ^ 1, kt + 1); ASYNC_WAIT(ASYNC_PER_CHUNK); }
    else                 { ASYNC_WAIT(0); }
    __syncthreads();
#pragma unroll
    for (int kk = 0; kk < BK; kk += 4) {
      float4 av[4], bv[4];
#pragma unroll
      for (int i = 0; i < 4; ++i) av[i] = *(const float4*)&As[buf][(m0 + i) * LDT + kk];
#pragma unroll
      for (int j = 0; j < 4; ++j) bv[j] = *(const float4*)&Bs[buf][(n0 + j) * LDT + kk];
#pragma unroll
      for (int i = 0; i < 4; ++i)
#pragma unroll
        for (int j = 0; j < 4; ++j)
          acc[i][j] += fminf(av[i].x, bv[j].x) + fminf(av[i].y, bv[j].y) +
                       fminf(av[i].z, bv[j].z) + fminf(av[i].w, bv[j].w);
    }
    __syncthreads();   // all waves done reading buf before it is refilled
    buf ^= 1;
  }

  float qs[4];
#pragma unroll
  for (int i = 0; i < 4; ++i) qs[i] = qsum[m0 + i];
  float4 e4 = *(const float4*)(esum + gn0 + n0);
  float es[4] = {e4.x, e4.y, e4.z, e4.w};
#pragma unroll
  for (int i = 0; i < 4; ++i) {
    float4 r;
    r.x = acc[i][0] / (qs[i] + es[0] - acc[i][0]);
    r.y = acc[i][1] / (qs[i] + es[1] - acc[i][1]);
    r.z = acc[i][2] / (qs[i] + es[2] - acc[i][2]);
    r.w = acc[i][3] / (qs[i] + es[3] - acc[i][3]);
    *(float4*)(jacc + (size_t)(m0 + i) * NPAD + gn0 + n0) = r;
  }
}

// ---------------- Kernel B: row sums (one wave32 per row) ----------------
__global__ __launch_bounds__(256) void row_sums(const float* __restrict__ q,
                                                const float* __restrict__ e,
                                                float* __restrict__ qsum,
                                                float* __restrict__ esum) {
  int row  = blockIdx.x * 8 + (threadIdx.x >> 5);
  int lane = threadIdx.x & 31;
  float s = 0.f;
  if (row < N_S) {
    const float4* p = (const float4*)(q + (size_t)row * DD);
    float4 a = p[lane], b = p[lane + 32];
    s = (a.x + a.y + a.z + a.w) + (b.x + b.y + b.z + b.w);
  } else {
    int er = row - N_S;
    if (er < N_E) {
      const float4* p = (const float4*)(e + (size_t)er * DD);
      float4 a = p[lane], b = p[lane + 32];
      s = (a.x + a.y + a.z + a.w) + (b.x + b.y + b.z + b.w);
    }
  }
#pragma unroll
  for (int off = 16; off > 0; off >>= 1) s += __shfl_xor(s, off, 32);
  if (lane == 0) {
    if (row < N_S) qsum[row] = s;
    else if (row - N_S < NPAD) esum[row - N_S] = s;  // zeros in padding
  }
}

// ---------------- Kernel C: top-10 / bottom-10 per sample ----------------
__device__ __forceinline__ bool better_top(float a, int ia, float b, int ib) {
  return (a > b) || (a == b && ia < ib);   // jax top_k tie-break: lower index
}
__device__ __forceinline__ bool better_bot(float a, int ia, float b, int ib) {
  return (a < b) || (a == b && ia < ib);
}

template <bool TOP>
__device__ void merge10(float* av, int* ai, const float* bv, const int* bi) {
  float ta[KSEL], tb[KSEL]; int ja[KSEL], jb[KSEL];
#pragma unroll
  for (int t = 0; t < KSEL; ++t) { ta[t] = av[t]; ja[t] = ai[t]; tb[t] = bv[t]; jb[t] = bi[t]; }
  int pa = 0, pb = 0;
#pragma unroll
  for (int t = 0; t < KSEL; ++t) {   // pa+pb == t <= 9, no overrun possible
    bool takeA = TOP ? better_top(ta[pa], ja[pa], tb[pb], jb[pb])
                     : better_bot(ta[pa], ja[pa], tb[pb], jb[pb]);
    if (takeA) { av[t] = ta[pa]; ai[t] = ja[pa]; ++pa; }
    else       { av[t] = tb[pb]; ai[t] = jb[pb]; ++pb; }
  }
}

__global__ __launch_bounds__(32) void topk10(const float* __restrict__ jacc,
                                             int* __restrict__ out) {
  const int s = blockIdx.x;
  const int lane = threadIdx.x;
  const float* row = jacc + (size_t)s * NPAD;
  float tv[KSEL], bv[KSEL]; int ti[KSEL], bi[KSEL];
#pragma unroll
  for (int t = 0; t < KSEL; ++t) {
    tv[t] = -__builtin_inff(); ti[t] = 0x7fffffff;
    bv[t] =  __builtin_inff(); bi[t] = 0x7fffffff;
  }
  for (int j = lane; j < N_E; j += 32) {
    float v = row[j];
    if (better_top(v, j, tv[KSEL - 1], ti[KSEL - 1])) {
      int p = KSEL - 1;
      while (p > 0 && better_top(v, j, tv[p - 1], ti[p - 1])) {
        tv[p] = tv[p - 1]; ti[p] = ti[p - 1]; --p;
      }
      tv[p] = v; ti[p] = j;
    }
    if (better_bot(v, j, bv[KSEL - 1], bi[KSEL - 1])) {
      int p = KSEL - 1;
      while (p > 0 && better_bot(v, j, bv[p - 1], bi[p - 1])) {
        bv[p] = bv[p - 1]; bi[p] = bi[p - 1]; --p;
      }
      bv[p] = v; bi[p] = j;
    }
  }
  __shared__ float sv[32][KSEL];  __shared__ int si[32][KSEL];
  __shared__ float sbv[32][KSEL]; __shared__ int sbi[32][KSEL];
#pragma unroll
  for (int t = 0; t < KSEL; ++t) {
    sv[lane][t] = tv[t]; si[lane][t] = ti[t];
    sbv[lane][t] = bv[t]; sbi[lane][t] = bi[t];
  }
  __syncthreads();
  for (int off = 16; off >= 1; off >>= 1) {
    if (lane < off) {
      merge10<true >(sv[lane],  si[lane],  sv[lane + off],  si[lane + off]);
      merge10<false>(sbv[lane], sbi[lane], sbv[lane + off], sbi[lane + off]);
    }
    __syncthreads();
  }
  if (lane == 0) {
#pragma unroll
    for (int t = 0; t < KSEL; ++t) {
      out[s * KSEL + t] = si[0][t];                       // top_idx
      out[N_S * KSEL + s * KSEL + t] = sbi[0][t];         // bot_idx
    }
  }
}

// ---------------- host entry ----------------
extern "C" void kernel_launch(void* const* d_in, const int* in_sizes, int n_in,
                              void* d_out, int out_size, void* d_ws, size_t ws_size,
                              hipStream_t stream) {
  const float* q = (const float*)d_in[0];   // [64, 4, 64] contiguous = [64, 256]
  const float* e = (const float*)d_in[1];   // [10000, 256]
  // d_in[2] is k == 10 (compile-time constant KSEL)

  float* ws   = (float*)d_ws;
  float* qsum = ws;                         // 64 floats
  float* esum = ws + N_S;                   // NPAD floats (16B-aligned: 256B offset)
  float* jacc = ws + N_S + NPAD;            // 64 * NPAD floats (16B-aligned offset)
  int*   out  = (int*)d_out;                // 64*10 top idx, then 64*10 bot idx

  const int sum_rows   = N_S + NPAD;        // 10112
  const int sum_blocks = sum_rows / 8;      // 1264 (8 waves per 256-thread block)
  const int gemm_blocks = NPAD / BN;        // 157

  row_sums<<<sum_blocks, 256, 0, stream>>>(q, e, qsum, esum);
  jaccard_minplus<<<gemm_blocks, 256, 0, stream>>>(q, e, qsum, esum, jacc);
  topk10<<<N_S, 32, 0, stream>>>(jacc, out);
}